// Blur_49976239456711
// MI455X (gfx1250) — compile-verified
//
#include <hip/hip_runtime.h>

// Blur: separable [1,3,3,1]/8 in each direction.
// Input  (8, 256, 64, 512) fp32, W wrap-pad 1, H reflect-pad 1.
// Output (8, 256, 63, 511) fp32.
//
// Bandwidth-bound: ~507 MB total HBM traffic, ~1 GFLOP -> ~22us floor at
// 23.3 TB/s. One block per (plane, 64-col strip); async DMA global->LDS
// staging (CDNA5 ASYNCcnt path: global_load_async_to_lds_b128/b32 +
// s_wait_asynccnt), 2-pass separable filter in LDS, non-temporal fp32 stores.

#define H_IN 64
#define W_IN 512
#define H_OUT 63
#define W_OUT 511
#define STRIP 64            // output columns per block
#define LDSW 72             // LDS row stride (floats): 288B = 16B-aligned rows, bank skew
#define NTHREADS 256

#if __has_builtin(__builtin_amdgcn_global_load_async_to_lds_b128) && \
    __has_builtin(__builtin_amdgcn_global_load_async_to_lds_b32)
#define USE_ASYNC_LDS 1
#else
#define USE_ASYNC_LDS 0
#endif

typedef int v4i __attribute__((vector_size(16)));
typedef __attribute__((address_space(1))) int as1_int;   // global ("__device__") AS
typedef __attribute__((address_space(3))) int as3_int;   // LDS ("__shared__") AS
typedef __attribute__((address_space(1))) v4i as1_v4i;
typedef __attribute__((address_space(3))) v4i as3_v4i;

__device__ __forceinline__ void wait_asynccnt0() {
#if __has_builtin(__builtin_amdgcn_s_wait_asynccnt)
    __builtin_amdgcn_s_wait_asynccnt(0);
#else
    asm volatile("s_wait_asynccnt 0" ::: "memory");
#endif
}

__global__ __launch_bounds__(NTHREADS) void blur4x4_kernel(
        const float* __restrict__ h, float* __restrict__ out) {
    // Tile columns: global x in [x0-1, x0+65], stored at LDS col (x - x0) + 4.
    __shared__ float s_in[H_IN][LDSW];     // staged input tile (67 used cols at [3..69])
    __shared__ float s_tmp[H_OUT][LDSW];   // vertically filtered tile

    const int strip = blockIdx.x;          // 0..7
    const int plane = blockIdx.y;          // 0..2047  (b*256 + c)
    const int x0    = strip * STRIP;
    const int tid   = threadIdx.x;

    const float* gplane = h + (size_t)plane * (H_IN * W_IN);

    // ---------------- Stage input tile into LDS ----------------
#if USE_ASYNC_LDS
    // Main body: 64 aligned columns per row as 16 x B128 DMAs (16B each).
    // 64 rows * 16 chunks = 1024 async ops; exactly 4 per thread, unrolled
    // so all DMAs issue back-to-back before the single asynccnt wait.
#pragma unroll
    for (int it = 0; it < (H_IN * 16) / NTHREADS; ++it) {
        const int idx = tid + it * NTHREADS;
        const int row = idx >> 4;
        const int k   = idx & 15;
        const float* src = gplane + row * W_IN + x0 + k * 4;
        float* dst = &s_in[row][4 + k * 4];          // byte off = row*288 + 16 + 16k (16B aligned)
        __builtin_amdgcn_global_load_async_to_lds_b128((as1_v4i*)src, (as3_v4i*)dst, 0, 0);
    }
    // Halo columns dx = -1, 64, 65 (wrap along W): 192 B32 DMAs, tid<192.
    if (tid < 3 * H_IN) {
        const int col = tid >> 6;                    // 0..2
        const int row = tid & 63;
        const int dx  = (col == 0) ? -1 : (63 + col);
        const int xg  = (x0 + dx) & (W_IN - 1);
        const float* src = gplane + row * W_IN + xg;
        float* dst = &s_in[row][4 + dx];
        __builtin_amdgcn_global_load_async_to_lds_b32((as1_int*)src, (as3_int*)dst, 0, 0);
    }
    wait_asynccnt0();
#else
    for (int idx = tid; idx < H_IN * 67; idx += NTHREADS) {
        const int row = idx / 67;
        const int dx  = (idx % 67) - 1;              // -1..65
        const int xg  = (x0 + dx) & (W_IN - 1);
        s_in[row][4 + dx] = gplane[row * W_IN + xg];
    }
#endif
    __syncthreads();

    const float w0 = 0.125f, w1 = 0.375f, w2 = 0.375f, w3 = 0.125f;

    // ---------------- Vertical pass (reflect pad) ----------------
    // tmp[r][x] = sum_i w[i] * orig[ymap(r+i)][x], ymap(0)=1, ymap(65)=62, else z-1.
    for (int idx = tid; idx < H_OUT * 67; idx += NTHREADS) {
        const int r  = idx / 67;
        const int c  = idx % 67;                     // dx = c-1  -> LDS col c+3
        const int lc = c + 3;
        const int y0 = (r == 0)  ? 1  : r - 1;       // ymap(r)
        const int y1 = r;                            // ymap(r+1)
        const int y2 = r + 1;                        // ymap(r+2)
        const int y3 = (r == 62) ? 62 : r + 2;       // ymap(r+3)
        s_tmp[r][lc] = w0 * s_in[y0][lc] + w1 * s_in[y1][lc]
                     + w2 * s_in[y2][lc] + w3 * s_in[y3][lc];
    }
    __syncthreads();

    // ---------------- Horizontal pass (wrap already staged) + NT store ----------------
    float* oplane = out + (size_t)plane * (H_OUT * W_OUT);
    for (int idx = tid; idx < H_OUT * STRIP; idx += NTHREADS) {
        const int r = idx >> 6;
        const int j = idx & 63;
        const int s = x0 + j;
        if (s < W_OUT) {
            // out[r][s] = sum_j' w[j'] * v[r][s-1+j'] ; LDS col = j + 3 + j'
            const float acc = w0 * s_tmp[r][j + 3] + w1 * s_tmp[r][j + 4]
                            + w2 * s_tmp[r][j + 5] + w3 * s_tmp[r][j + 6];
            __builtin_nontemporal_store(acc, &oplane[r * W_OUT + s]);
        }
    }
}

extern "C" void kernel_launch(void* const* d_in, const int* in_sizes, int n_in,
                              void* d_out, int out_size, void* d_ws, size_t ws_size,
                              hipStream_t stream) {
    const float* h = (const float*)d_in[0];   // (8,256,64,512) fp32
    float* out = (float*)d_out;               // (8,256,63,511) fp32
    (void)in_sizes; (void)n_in; (void)out_size; (void)d_ws; (void)ws_size;
    dim3 grid(W_IN / STRIP, 8 * 256);         // (8 strips, 2048 planes)
    blur4x4_kernel<<<grid, NTHREADS, 0, stream>>>(h, out);
}